// VoxelSetAbstraction_6597069767108
// MI455X (gfx1250) — compile-verified
//
#include <hip/hip_runtime.h>
#include <hip/hip_bf16.h>

// CDNA5 / gfx1250. wave32. WMMA f32 16x16x4 for the MLPs + fusion GEMM.

typedef __attribute__((ext_vector_type(2))) float v2f;
typedef __attribute__((ext_vector_type(8))) float v8f;

#define NPTS   16384
#define NKP    2048
#define NBATCH 2
#define CBEV   256
#define HBEV   200
#define WBEV   176
#define FDIM   288
#define ODIM   128
#define EPSBN  1e-5f

// ---------------------------------------------------------------------------
// 1) Farthest point sampling: 1 workgroup (32 waves) per batch.
//    Point coords live in registers (16 pts/thread); per-iteration block
//    argmax via wave32 shfl reduction + LDS combine. Matches jnp.argmax
//    tie-break (lowest index wins).
// ---------------------------------------------------------------------------
__global__ __launch_bounds__(1024) void fps_kernel(const float4* __restrict__ pts,
                                                   float4* __restrict__ kp)
{
    __shared__ float redv[32];
    __shared__ int   redi[32];
    __shared__ int   s_best;

    const int b    = blockIdx.x;
    const int tid  = threadIdx.x;
    const int lane = tid & 31;
    const int wv   = tid >> 5;
    const float4* p = pts + (size_t)b * NPTS;

    float px[16], py[16], pz[16], mind[16];
#pragma unroll
    for (int j = 0; j < 16; ++j) {
        float4 q = p[tid + j * 1024];
        px[j] = q.x; py[j] = q.y; pz[j] = q.z;
        mind[j] = 1e10f;
    }

    int cur = 0;
    if (tid == 0) kp[(size_t)b * NKP + 0] = p[0];

    for (int it = 1; it < NKP; ++it) {
        float4 lp = p[cur];                      // broadcast load (L2-hot)
        float bv = -1.0f; int bi = 0;
#pragma unroll
        for (int j = 0; j < 16; ++j) {
            float dx = px[j] - lp.x;
            float dy = py[j] - lp.y;
            float dz = pz[j] - lp.z;
            float d  = dx * dx + dy * dy + dz * dz;
            mind[j]  = fminf(mind[j], d);
            int idx  = tid + j * 1024;
            if (mind[j] > bv) { bv = mind[j]; bi = idx; }   // ascending idx: first-max kept
        }
        // wave32 reduction, prefer smaller index on ties
#pragma unroll
        for (int off = 16; off > 0; off >>= 1) {
            float ov = __shfl_down(bv, off, 32);
            int   oi = __shfl_down(bi, off, 32);
            if (ov > bv || (ov == bv && oi < bi)) { bv = ov; bi = oi; }
        }
        if (lane == 0) { redv[wv] = bv; redi[wv] = bi; }
        __syncthreads();
        if (wv == 0) {
            float v2 = redv[lane]; int i2 = redi[lane];
#pragma unroll
            for (int off = 16; off > 0; off >>= 1) {
                float ov = __shfl_down(v2, off, 32);
                int   oi = __shfl_down(i2, off, 32);
                if (ov > v2 || (ov == v2 && oi < i2)) { v2 = ov; i2 = oi; }
            }
            if (lane == 0) s_best = i2;
        }
        __syncthreads();
        cur = s_best;
        if (tid == 0) kp[(size_t)b * NKP + it] = p[cur];
    }
}

// ---------------------------------------------------------------------------
// 2) Bilinear BEV sampling: one block per keypoint, one thread per channel.
//    Writes F[:, 0:256].
// ---------------------------------------------------------------------------
__global__ __launch_bounds__(256) void bilinear_kernel(const float* __restrict__ bev,
                                                       const float4* __restrict__ kp,
                                                       float* __restrict__ F)
{
    const int g = blockIdx.x;                // b*NKP + m
    const int b = g / NKP;
    const int c = threadIdx.x;               // 0..255

    float4 k = kp[g];
    float xi = (k.x - 0.0f)    / 0.05f / 8.0f;
    float yi = (k.y - -40.0f)  / 0.05f / 8.0f;

    float x0f = floorf(xi), y0f = floorf(yi);
    int x0 = min(max((int)x0f,     0), WBEV - 1);
    int x1 = min(max((int)x0f + 1, 0), WBEV - 1);
    int y0 = min(max((int)y0f,     0), HBEV - 1);
    int y1 = min(max((int)y0f + 1, 0), HBEV - 1);

    float x0t = (float)x0, x1t = (float)x1, y0t = (float)y0, y1t = (float)y1;
    float wa = (x1t - xi) * (y1t - yi);
    float wb = (x1t - xi) * (yi - y0t);
    float wc = (xi - x0t) * (y1t - yi);
    float wd = (xi - x0t) * (yi - y0t);

    const float* im = bev + ((size_t)b * CBEV + c) * (HBEV * WBEV);
    float Ia = im[y0 * WBEV + x0];
    float Ib = im[y1 * WBEV + x0];
    float Ic = im[y0 * WBEV + x1];
    float Id = im[y1 * WBEV + x1];

    F[(size_t)g * FDIM + c] = Ia * wa + Ib * wb + Ic * wc + Id * wd;
}

// ---------------------------------------------------------------------------
// 3) 16-NN per keypoint (ascending d2, stable): one thread per keypoint.
// ---------------------------------------------------------------------------
__global__ __launch_bounds__(128) void knn_kernel(const float4* __restrict__ pts,
                                                  const float4* __restrict__ kp,
                                                  float* __restrict__ nnd,
                                                  int* __restrict__ nni)
{
    int g = blockIdx.x * blockDim.x + threadIdx.x;
    if (g >= NBATCH * NKP) return;
    int b = g / NKP;
    float4 k = kp[g];
    const float4* p = pts + (size_t)b * NPTS;

    float nd[16]; int ni[16];
#pragma unroll
    for (int t = 0; t < 16; ++t) { nd[t] = 3.4e38f; ni[t] = 0; }

    for (int j = 0; j < NPTS; ++j) {
        float4 q = p[j];
        float dx = q.x - k.x, dy = q.y - k.y, dz = q.z - k.z;
        float d = dx * dx + dy * dy + dz * dz;
        if (d < nd[15]) {
            float cd = d; int ci = j;
#pragma unroll
            for (int t = 0; t < 16; ++t) {
                if (cd < nd[t]) {
                    float tv = nd[t]; int ti = ni[t];
                    nd[t] = cd; ni[t] = ci; cd = tv; ci = ti;
                }
            }
        }
    }
#pragma unroll
    for (int t = 0; t < 16; ++t) {
        nnd[(size_t)g * 16 + t] = nd[t];
        nni[(size_t)g * 16 + t] = ni[t];
    }
}

// ---------------------------------------------------------------------------
// 4) Grouping + 2-layer MLP + maxpool, one wave32 per keypoint per scale.
//    MLP1: (16 samples x 4) @ (4 x 16)  -> one V_WMMA_F32_16X16X4_F32
//    MLP2: (16 x 16) @ (16 x 16)        -> 4 chained WMMAs (K split by 4),
//    with an LDS round trip to restripe the D layout into the A layout.
//    Writes F[:, 256 + 16*scale : 272 + 16*scale].
// ---------------------------------------------------------------------------
__global__ __launch_bounds__(32) void group_mlp_kernel(
    const float4* __restrict__ pts, const float4* __restrict__ kp,
    const float* __restrict__ nnd, const int* __restrict__ nni,
    const float* __restrict__ W1,
    const float* __restrict__ ga1, const float* __restrict__ be1,
    const float* __restrict__ mu1, const float* __restrict__ va1,
    const float* __restrict__ W2,
    const float* __restrict__ ga2, const float* __restrict__ be2,
    const float* __restrict__ mu2, const float* __restrict__ va2,
    float radius2, int scale, float* __restrict__ F)
{
    __shared__ float hmat[16][17];           // padded to dodge bank conflicts

    const int g     = blockIdx.x;            // b*NKP + m
    const int b     = g / NKP;
    const int lane  = threadIdx.x;
    const int n     = lane & 15;             // column (N) / row-of-A (M)
    const int khalf = lane >> 4;

    float4 kpt = kp[g];

    // neighbor features g = [dx, dy, dz, intensity] for sample row n
    float d2 = nnd[(size_t)g * 16 + n];
    int   i0 = nni[(size_t)g * 16 + 0];
    int   ii = (d2 <= radius2) ? nni[(size_t)g * 16 + n] : i0;
    float4 q = pts[(size_t)b * NPTS + ii];
    float f0 = q.x - kpt.x;
    float f1 = q.y - kpt.y;
    float f2 = q.z - kpt.z;
    float f3 = q.w;

    // A (16x4): lanes 0-15 carry K={0,1}, lanes 16-31 carry K={2,3}
    v2f a;
    a.x = khalf ? f2 : f0;
    a.y = khalf ? f3 : f1;
    // B (4x16): VGPR j / lane-half h holds K = 2h + j, N = lane%16
    v2f w;
    w.x = W1[(2 * khalf + 0) * 16 + n];
    w.y = W1[(2 * khalf + 1) * 16 + n];

    v8f c = {0.f, 0.f, 0.f, 0.f, 0.f, 0.f, 0.f, 0.f};
    c = __builtin_amdgcn_wmma_f32_16x16x4_f32(false, a, false, w,
                                              (short)0, c, false, false);

    // BN1 + ReLU (per hidden channel n), spill D layout to LDS as 16x16
    float sc1 = ga1[n] * rsqrtf(va1[n] + EPSBN);
    float bi1 = be1[n] - mu1[n] * sc1;
#pragma unroll
    for (int t = 0; t < 8; ++t) {
        float h = fmaxf(c[t] * sc1 + bi1, 0.0f);   // D: M = t + 8*khalf, N = n
        hmat[t + 8 * khalf][n] = h;
    }
    __syncthreads();
    asm volatile("s_wait_dscnt 0" ::: "memory");

    // MLP2: h(16x16) @ W2(16x16), K tiled by 4
    v8f c2 = {0.f, 0.f, 0.f, 0.f, 0.f, 0.f, 0.f, 0.f};
#pragma unroll
    for (int kk = 0; kk < 4; ++kk) {
        int k0 = 4 * kk + 2 * khalf;
        v2f a2; a2.x = hmat[n][k0]; a2.y = hmat[n][k0 + 1];
        v2f w2; w2.x = W2[k0 * 16 + n]; w2.y = W2[(k0 + 1) * 16 + n];
        c2 = __builtin_amdgcn_wmma_f32_16x16x4_f32(false, a2, false, w2,
                                                   (short)0, c2, false, false);
    }

    // BN2 + ReLU + max over 16 samples (8 in-lane rows + cross-half shfl)
    float sc2 = ga2[n] * rsqrtf(va2[n] + EPSBN);
    float bi2 = be2[n] - mu2[n] * sc2;
    float mx = -3.4e38f;
#pragma unroll
    for (int t = 0; t < 8; ++t)
        mx = fmaxf(mx, fmaxf(c2[t] * sc2 + bi2, 0.0f));
    float om = __shfl_xor(mx, 16, 32);
    mx = fmaxf(mx, om);

    if (lane < 16)
        F[(size_t)g * FDIM + 256 + scale * 16 + n] = mx;
}

// ---------------------------------------------------------------------------
// 5) Fusion GEMM: (4096 x 288) @ (288 x 128) + BN + ReLU.
//    One wave32 per 16x16 output tile; 72 V_WMMA_F32_16X16X4_F32 per tile.
// ---------------------------------------------------------------------------
__global__ __launch_bounds__(256) void fusion_kernel(
    const float* __restrict__ F, const float* __restrict__ W,
    const float* __restrict__ ga, const float* __restrict__ be,
    const float* __restrict__ mu, const float* __restrict__ va,
    float* __restrict__ out)
{
    const int wv    = threadIdx.x >> 5;
    const int lane  = threadIdx.x & 31;
    const int tile  = blockIdx.x * 8 + wv;     // 2048 tiles = (4096/16)*(128/16)
    const int tm    = tile >> 3;
    const int tn    = tile & 7;
    const int n     = lane & 15;
    const int khalf = lane >> 4;

    const int rowA = tm * 16 + n;              // A row (M = lane%16)
    const int ncol = tn * 16 + n;              // output column
    const float* arow = F + (size_t)rowA * FDIM;

    v8f c = {0.f, 0.f, 0.f, 0.f, 0.f, 0.f, 0.f, 0.f};
#pragma unroll 4
    for (int k0 = 0; k0 < FDIM; k0 += 4) {
        v2f a;
        a.x = arow[k0 + 2 * khalf + 0];
        a.y = arow[k0 + 2 * khalf + 1];
        v2f w;
        w.x = W[(size_t)(k0 + 2 * khalf + 0) * ODIM + ncol];
        w.y = W[(size_t)(k0 + 2 * khalf + 1) * ODIM + ncol];
        c = __builtin_amdgcn_wmma_f32_16x16x4_f32(false, a, false, w,
                                                  (short)0, c, false, false);
    }

    float sc = ga[ncol] * rsqrtf(va[ncol] + EPSBN);
    float bi = be[ncol] - mu[ncol] * sc;
#pragma unroll
    for (int t = 0; t < 8; ++t) {
        int m = tm * 16 + t + 8 * khalf;       // D: M = vgpr + 8*half
        out[(size_t)m * ODIM + ncol] = fmaxf(c[t] * sc + bi, 0.0f);
    }
}

// ---------------------------------------------------------------------------
// Host-side launch.  Input flattening (setup_inputs dict order, bn tuples
// expanded (gamma, beta, mean, var)):
//   0: points (2*16384*4)        1: bev_features (2*256*200*176)
//   2: s0_W1   3-6: s0_bn1       7: s0_W2   8-11: s0_bn2
//  12: s1_W1  13-16: s1_bn1     17: s1_W2  18-21: s1_bn2
//  22: fus_W  23-26: fus_bn
// ---------------------------------------------------------------------------
extern "C" void kernel_launch(void* const* d_in, const int* in_sizes, int n_in,
                              void* d_out, int out_size, void* d_ws, size_t ws_size,
                              hipStream_t stream)
{
    const float4* pts = (const float4*)d_in[0];
    const float*  bev = (const float*)d_in[1];

    const float* s0_W1 = (const float*)d_in[2];
    const float* s0_g1 = (const float*)d_in[3];
    const float* s0_b1 = (const float*)d_in[4];
    const float* s0_m1 = (const float*)d_in[5];
    const float* s0_v1 = (const float*)d_in[6];
    const float* s0_W2 = (const float*)d_in[7];
    const float* s0_g2 = (const float*)d_in[8];
    const float* s0_b2 = (const float*)d_in[9];
    const float* s0_m2 = (const float*)d_in[10];
    const float* s0_v2 = (const float*)d_in[11];

    const float* s1_W1 = (const float*)d_in[12];
    const float* s1_g1 = (const float*)d_in[13];
    const float* s1_b1 = (const float*)d_in[14];
    const float* s1_m1 = (const float*)d_in[15];
    const float* s1_v1 = (const float*)d_in[16];
    const float* s1_W2 = (const float*)d_in[17];
    const float* s1_g2 = (const float*)d_in[18];
    const float* s1_b2 = (const float*)d_in[19];
    const float* s1_m2 = (const float*)d_in[20];
    const float* s1_v2 = (const float*)d_in[21];

    const float* fus_W = (const float*)d_in[22];
    const float* fus_g = (const float*)d_in[23];
    const float* fus_b = (const float*)d_in[24];
    const float* fus_m = (const float*)d_in[25];
    const float* fus_v = (const float*)d_in[26];

    float* out = (float*)d_out;

    // workspace carve-up
    char* ws = (char*)d_ws;
    float4* kp  = (float4*)ws;                       ws += (size_t)NBATCH * NKP * sizeof(float4);
    float*  nnd = (float*)ws;                        ws += (size_t)NBATCH * NKP * 16 * sizeof(float);
    int*    nni = (int*)ws;                          ws += (size_t)NBATCH * NKP * 16 * sizeof(int);
    float*  F   = (float*)ws;                        // (NBATCH*NKP) x 288

    fps_kernel<<<NBATCH, 1024, 0, stream>>>(pts, kp);
    bilinear_kernel<<<NBATCH * NKP, 256, 0, stream>>>(bev, kp, F);
    knn_kernel<<<(NBATCH * NKP) / 128, 128, 0, stream>>>(pts, kp, nnd, nni);
    group_mlp_kernel<<<NBATCH * NKP, 32, 0, stream>>>(
        pts, kp, nnd, nni,
        s0_W1, s0_g1, s0_b1, s0_m1, s0_v1,
        s0_W2, s0_g2, s0_b2, s0_m2, s0_v2,
        0.4f * 0.4f, 0, F);
    group_mlp_kernel<<<NBATCH * NKP, 32, 0, stream>>>(
        pts, kp, nnd, nni,
        s1_W1, s1_g1, s1_b1, s1_m1, s1_v1,
        s1_W2, s1_g2, s1_b2, s1_m2, s1_v2,
        0.8f * 0.8f, 1, F);
    fusion_kernel<<<(NBATCH * NKP) / 16 * (ODIM / 16) / 8, 256, 0, stream>>>(
        F, fus_W, fus_g, fus_b, fus_m, fus_v, out);
}